// SpatialTransformer_47665547051459
// MI455X (gfx1250) — compile-verified
//
#include <hip/hip_runtime.h>
#include <stdint.h>

// Spatial transformer (rotation-only) with bilinear sampling.
// B=64, H=W=512, C=3, fp32. HBM-roofline bound (~402 MB -> ~17us @ 23.3 TB/s).
// Strategy: per 32x32 output tile, TDM-load the rotated source bbox (<=48x48x3
// floats = 27.6 KB) into LDS with tensor_load_to_lds, then do all bilinear
// gathers out of LDS. Global memory sees only contiguous row reads + coalesced
// writes; the divergent gather traffic never leaves the WGP.

#define OUT_H 512
#define OUT_W 512
#define NCH   3
#define TILE  32
#define BMAX  48      // 31*sqrt(2) ~ 43.8 source span + floor/bilinear + margin
#define BLOCK 256
#define BATCH 64

typedef uint32_t cd5_u32x4 __attribute__((ext_vector_type(4)));
typedef int      cd5_i32x8 __attribute__((ext_vector_type(8)));
typedef int      cd5_i32x4 __attribute__((ext_vector_type(4)));

__device__ __forceinline__ int clampi(int v, int lo, int hi) {
  v = v < lo ? lo : v;
  return v > hi ? hi : v;
}

__global__ __launch_bounds__(BLOCK) void st_rot_bilinear(
    const float* __restrict__ U, const float* __restrict__ theta,
    float* __restrict__ out) {
  __shared__ float smem[BMAX * BMAX * NCH];  // 27648 bytes

  const int tiles_x = OUT_W / TILE;
  const int tx0 = ((int)blockIdx.x % tiles_x) * TILE;
  const int ty0 = ((int)blockIdx.x / tiles_x) * TILE;
  const int b   = (int)blockIdx.y;

  const float th = theta[b];          // uniform per block
  const float ct = cosf(th);
  const float st = sinf(th);

  const float gsx = 2.0f / (float)(OUT_W - 1);
  const float gsy = 2.0f / (float)(OUT_H - 1);
  const float hx  = (float)(OUT_W - 1) * 0.5f;
  const float hy  = (float)(OUT_H - 1) * 0.5f;

  // Same formula for bbox corners and per-pixel sampling (monotone FP ops);
  // +/-1 px margin below covers any rounding asymmetry.
#define SRC_X(gx, gy) ((ct * (gx) - st * (gy) + 1.0f) * hx)
#define SRC_Y(gx, gy) ((st * (gx) + ct * (gy) + 1.0f) * hy)

  const float gxa = (float)tx0 * gsx - 1.0f;
  const float gxb = (float)(tx0 + TILE - 1) * gsx - 1.0f;
  const float gya = (float)ty0 * gsy - 1.0f;
  const float gyb = (float)(ty0 + TILE - 1) * gsy - 1.0f;

  const float x00 = SRC_X(gxa, gya), x10 = SRC_X(gxb, gya);
  const float x01 = SRC_X(gxa, gyb), x11 = SRC_X(gxb, gyb);
  const float y00 = SRC_Y(gxa, gya), y10 = SRC_Y(gxb, gya);
  const float y01 = SRC_Y(gxa, gyb), y11 = SRC_Y(gxb, gyb);

  const float xmin = fminf(fminf(x00, x10), fminf(x01, x11));
  const float xmax = fmaxf(fmaxf(x00, x10), fmaxf(x01, x11));
  const float ymin = fminf(fminf(y00, y10), fminf(y01, y11));
  const float ymax = fmaxf(fmaxf(y00, y10), fmaxf(y01, y11));

  // Source bbox (clamped to image, with margin); all clamped bilinear corners
  // for this tile land inside [bx0,bx1]x[by0,by1].
  const int bx0 = clampi((int)floorf(xmin) - 1, 0, OUT_W - 1);
  const int bx1 = clampi((int)floorf(xmax) + 2, bx0, OUT_W - 1);
  const int by0 = clampi((int)floorf(ymin) - 1, 0, OUT_H - 1);
  const int by1 = clampi((int)floorf(ymax) + 2, by0, OUT_H - 1);
  const int bw  = bx1 - bx0 + 1;   // <= 48
  const int bh  = by1 - by0 + 1;   // <= 48

  // ---- TDM: async DMA of the bbox (bh rows x bw*3 floats, row stride W*3)
  // into LDS. One descriptor, issued by wave 0 only (EXEC-independent).
  if (threadIdx.x < 32) {
    const uint64_t gaddr = (uint64_t)(uintptr_t)U +
        (uint64_t)(uint32_t)(((b * OUT_H + by0) * OUT_W + bx0) * NCH) * 4ull;
    const uint32_t lds_base = (uint32_t)(uintptr_t)&smem[0];
    const uint32_t tile0   = (uint32_t)(bw * NCH);   // elems along dim0 (<=144)
    const uint32_t tile1   = (uint32_t)bh;           // rows
    const uint32_t stride0 = OUT_W * NCH;            // 1536 elems

#define RFL(x) ((uint32_t)__builtin_amdgcn_readfirstlane((int)(x)))
    // D# group 0: count=1 | lds_addr | global_addr[56:0] | type=2
    cd5_u32x4 g0 = {
        RFL(0x1u),
        RFL(lds_base),
        RFL((uint32_t)gaddr),
        RFL(((uint32_t)(gaddr >> 32) & 0x01FFFFFFu) | 0x80000000u)};
    // D# group 1: data_size=4B; tensor_dim0/1 == tile_dim0/1 (bbox pre-clamped
    // in-image, so no OOB); dim0 stride = W*3; no padding/iterate/multicast.
    cd5_i32x8 g1 = {
        (int)RFL(0x2u << 16),                                  // data_size=2 (4B)
        (int)RFL((tile0 & 0xFFFFu) << 16),                     // tdim0[15:0]
        (int)RFL((tile0 >> 16) | ((tile1 & 0xFFFFu) << 16)),   // tdim0[31:16]|tdim1[15:0]
        (int)RFL((tile1 >> 16) | ((tile0 & 0xFFFFu) << 16)),   // tdim1[31:16]|tile_dim0
        (int)RFL(tile1 & 0xFFFFu),                             // tile_dim1, tile_dim2=0
        (int)RFL(stride0),                                     // tdim0_stride[31:0]
        0,                                                     // stride hi bits
        0};
    cd5_i32x4 gz4  = {0, 0, 0, 0};
    cd5_i32x8 gz8  = {0, 0, 0, 0, 0, 0, 0, 0};
    // clang-23 / therock-10.0 form: 6 args (g0, g1, g2, g3, extra, cpol)
    __builtin_amdgcn_tensor_load_to_lds(g0, g1, gz4, gz4, gz8, 0);
    __builtin_amdgcn_s_wait_tensorcnt(0);
#undef RFL
  }
  // Let the optimizer know LDS was written behind its back by the TDM.
  {
    float* sp = &smem[0];
    asm volatile("" : "+v"(sp)::"memory");
  }
  __syncthreads();

  // ---- Bilinear sampling out of LDS; 4 pixels/thread, coalesced stores.
  for (int i = (int)threadIdx.x; i < TILE * TILE; i += BLOCK) {
    const int px = tx0 + (i & (TILE - 1));
    const int py = ty0 + (i >> 5);

    const float gx = (float)px * gsx - 1.0f;
    const float gy = (float)py * gsy - 1.0f;
    const float x  = SRC_X(gx, gy);
    const float y  = SRC_Y(gx, gy);

    const float xf = floorf(x);
    const float yf = floorf(y);
    const int x0 = clampi((int)xf, 0, OUT_W - 1);
    const int x1 = clampi((int)xf + 1, 0, OUT_W - 1);
    const int y0 = clampi((int)yf, 0, OUT_H - 1);
    const int y1 = clampi((int)yf + 1, 0, OUT_H - 1);

    // Weights from *clamped* integer corners, exactly as the reference.
    const float wa = ((float)x1 - x) * ((float)y1 - y);
    const float wb = ((float)x1 - x) * (y - (float)y0);
    const float wc = (x - (float)x0) * ((float)y1 - y);
    const float wd = (x - (float)x0) * (y - (float)y0);

    const int lx0 = clampi(x0 - bx0, 0, bw - 1);
    const int lx1 = clampi(x1 - bx0, 0, bw - 1);
    const int ly0 = clampi(y0 - by0, 0, bh - 1);
    const int ly1 = clampi(y1 - by0, 0, bh - 1);

    const int ia = (ly0 * bw + lx0) * NCH;
    const int ib = (ly1 * bw + lx0) * NCH;
    const int ic = (ly0 * bw + lx1) * NCH;
    const int id = (ly1 * bw + lx1) * NCH;

    const size_t o = (((size_t)b * OUT_H + py) * (size_t)OUT_W + px) * NCH;
    out[o + 0] = wa * smem[ia + 0] + wb * smem[ib + 0] + wc * smem[ic + 0] + wd * smem[id + 0];
    out[o + 1] = wa * smem[ia + 1] + wb * smem[ib + 1] + wc * smem[ic + 1] + wd * smem[id + 1];
    out[o + 2] = wa * smem[ia + 2] + wb * smem[ib + 2] + wc * smem[ic + 2] + wd * smem[id + 2];
  }
#undef SRC_X
#undef SRC_Y
}

extern "C" void kernel_launch(void* const* d_in, const int* in_sizes, int n_in,
                              void* d_out, int out_size, void* d_ws, size_t ws_size,
                              hipStream_t stream) {
  const float* U     = (const float*)d_in[0];   // [64, 512, 512, 3] fp32
  const float* theta = (const float*)d_in[1];   // [64, 1] fp32
  float* out = (float*)d_out;                   // [64, 512, 512, 3] fp32

  dim3 grid((OUT_W / TILE) * (OUT_H / TILE), BATCH);
  st_rot_bilinear<<<grid, BLOCK, 0, stream>>>(U, theta, out);

  (void)in_sizes; (void)n_in; (void)out_size; (void)d_ws; (void)ws_size;
}